// DynamicQAgent_33681133535602
// MI455X (gfx1250) — compile-verified
//
#include <hip/hip_runtime.h>
#include <stdint.h>
#include <utility>

// DynamicQAgent scan: 8192 sessions x 1000 trials, sequential in t, parallel in s.
// Memory-bound (~164 MB @ 23.3 TB/s ≈ 7 us). Per-thread session recurrence with a
// time-chunked, double-buffered async DMA pipeline through LDS using CDNA5
// GLOBAL_LOAD_ASYNC_TO_LDS / GLOBAL_STORE_ASYNC_FROM_LDS (ASYNCcnt tracked).
// The instruction immediate offset is added to BOTH the LDS and global address
// (CDNA5 ISA), so each burst needs only one VGPR address pair +
// one LDS-offset VGPR; all intra-chunk stepping is folded into offset: imms.

#define N_SESS   8192
#define N_TRIALS 1000
#define BLOCK    64
#define CHUNK    40                       // 25 chunks, no tail
#define NCHUNK   (N_TRIALS / CHUNK)

#define IN_BYTES   (CHUNK * 3 * 4)        // 480 B / thread / chunk (16B aligned)
#define OUT_BYTES  (CHUNK * 2 * 4)        // 320 B / thread / chunk (16B aligned)
#define IN_LOADS   (IN_BYTES  / 16)       // 30 x b128
#define OUT_STORES (OUT_BYTES / 16)       // 20 x b128  (ASYNCcnt peak 50 < 63)

template <int OFF>
__device__ __forceinline__ void async_load_b128(uint32_t lds_byte_off, const void* gaddr) {
  // VDST = LDS byte address, VADDR = 64-bit global address (GV mode);
  // offset: imm is applied to both LDS and global addresses.
  asm volatile("global_load_async_to_lds_b128 %0, %1, off offset:%c2"
               :: "v"(lds_byte_off), "v"(gaddr), "i"(OFF) : "memory");
}
template <int OFF>
__device__ __forceinline__ void async_store_b128(const void* gaddr, uint32_t lds_byte_off) {
  asm volatile("global_store_async_from_lds_b128 %0, %1, off offset:%c2"
               :: "v"(gaddr), "v"(lds_byte_off), "i"(OFF) : "memory");
}

template <int... Is>
__device__ __forceinline__ void load_row_impl(uint32_t lds, const void* g,
                                              std::integer_sequence<int, Is...>) {
  (async_load_b128<Is * 16>(lds, g), ...);
}
__device__ __forceinline__ void load_row(uint32_t lds, const void* g) {
  load_row_impl(lds, g, std::make_integer_sequence<int, IN_LOADS>{});
}

template <int... Is>
__device__ __forceinline__ void store_row_impl(const void* g, uint32_t lds,
                                               std::integer_sequence<int, Is...>) {
  (async_store_b128<Is * 16>(g, lds), ...);
}
__device__ __forceinline__ void store_row(const void* g, uint32_t lds) {
  store_row_impl(g, lds, std::make_integer_sequence<int, OUT_STORES>{});
}

__device__ __forceinline__ void wait_async0() {
  asm volatile("s_wait_asynccnt 0x0" ::: "memory");
}
__device__ __forceinline__ void wait_ds0() {
  asm volatile("s_wait_dscnt 0x0" ::: "memory");
}

__global__ __launch_bounds__(BLOCK)
void DynamicQAgent_kernel(const float* __restrict__ inp,
                          const float* __restrict__ alphas,
                          const float* __restrict__ gammas2,
                          const float* __restrict__ kvals,
                          float* __restrict__ out)
{
  // Double-buffered staging: 2*30KB in + 2*20KB out = 100KB (< 320KB/WGP)
  __shared__ __align__(16) float s_in [2][BLOCK * CHUNK * 3];
  __shared__ __align__(16) float s_out[2][BLOCK * CHUNK * 2];

  const int tid  = threadIdx.x;
  const int sess = blockIdx.x * BLOCK + tid;

  // Uniform params -> scalar loads
  const float a0 = alphas[0], a1 = alphas[1], a2 = alphas[2], a3 = alphas[3];
  const float g0 = gammas2[0], g1 = gammas2[1];
  const float k0 = kvals[0], k1 = kvals[1], k2 = kvals[2], k3 = kvals[3];

  const char* gin  = (const char*)(inp + (size_t)sess * (N_TRIALS * 3));
  char*       gout = (char*)(out + (size_t)sess * (N_TRIALS * 2));

  // LDS byte offsets of this thread's private rows (ptrtoint also escapes the
  // arrays so DSE keeps the ds stores feeding the async stores).
  uint32_t in_off[2], out_off[2];
  in_off[0]  = (uint32_t)(uintptr_t)&s_in [0][tid * CHUNK * 3];
  in_off[1]  = (uint32_t)(uintptr_t)&s_in [1][tid * CHUNK * 3];
  out_off[0] = (uint32_t)(uintptr_t)&s_out[0][tid * CHUNK * 2];
  out_off[1] = (uint32_t)(uintptr_t)&s_out[1][tid * CHUNK * 2];

  // Prologue: async-load chunk 0
  load_row(in_off[0], gin);

  float sL = 0.0f, sR = 0.0f, lam = 0.5f;

  for (int k = 0; k < NCHUNK; ++k) {
    const int cur = k & 1, nxt = cur ^ 1;

    // Loads(k) done; stores(k-2) done -> buffers 'cur' are safe to use/overwrite.
    wait_async0();

    // Drain chunk k-1 results (overlaps compute of chunk k)
    if (k)
      store_row(gout + (size_t)(k - 1) * OUT_BYTES, out_off[nxt]);

    // Prefetch chunk k+1 (overlaps compute of chunk k)
    if (k + 1 < NCHUNK)
      load_row(in_off[nxt], gin + (size_t)(k + 1) * IN_BYTES);

    // Compute chunk k: sel_left is one-hot -> the whole step is selects+FMAs.
    const float* rin  = &s_in [cur][tid * CHUNK * 3];
    float*       rout = &s_out[cur][tid * CHUNK * 2];
    #pragma unroll 8
    for (int tt = 0; tt < CHUNK; ++tt) {
      const float cL = rin[tt * 3 + 0];
      const float o  = rin[tt * 3 + 2];
      const bool left = cL > 0.5f;
      const bool win  = o  > 0.5f;
      // j = left?(win?0:1):(win?2:3); jr = (j+2)&3
      const float kj  = left ? (win ? k0 : k1) : (win ? k2 : k3);
      const float aj  = left ? (win ? a0 : a1) : (win ? a2 : a3);
      const float kjr = left ? (win ? k2 : k3) : (win ? k0 : k1);
      const float ajr = left ? (win ? a2 : a3) : (win ? a0 : a1);
      const float ld  = aj  * (kj  - sL) * lam;     // old lam, old state
      const float rd  = ajr * (kjr - sR) * lam;
      // global_diffs[j]: 0->|k0-sL| 1->|k1-sL| 2->|k0-sR| 3->|k1-sR|
      const float gd  = left ? (win ? fabsf(k0 - sL) : fabsf(k1 - sL))
                             : (win ? fabsf(k0 - sR) : fabsf(k1 - sR));
      const float gg  = win ? g0 : g1;              // gammas = [g0,g1,g0,g1]
      lam += gg * (gd - lam);
      sL  += ld;
      sR  += rd;
      rout[tt * 2 + 0] = sL;
      rout[tt * 2 + 1] = sR;
    }
    // LDS writes must land before the async-store engine reads them next iter.
    wait_ds0();
  }

  // Epilogue: drain final chunk
  store_row(gout + (size_t)(NCHUNK - 1) * OUT_BYTES, out_off[(NCHUNK - 1) & 1]);
  wait_async0();
}

extern "C" void kernel_launch(void* const* d_in, const int* in_sizes, int n_in,
                              void* d_out, int out_size, void* d_ws, size_t ws_size,
                              hipStream_t stream) {
  const float* inp     = (const float*)d_in[0];  // (8192, 1000, 3)
  const float* alphas  = (const float*)d_in[1];  // (4,)
  const float* gammas2 = (const float*)d_in[2];  // (2,)
  const float* kvals   = (const float*)d_in[3];  // (4,)
  float*       out     = (float*)d_out;          // (8192, 1000, 2)

  dim3 grid(N_SESS / BLOCK);
  dim3 block(BLOCK);
  hipLaunchKernelGGL(DynamicQAgent_kernel, grid, block, 0, stream,
                     inp, alphas, gammas2, kvals, out);
}